// OldTensorProductConvLayer_58145267253788
// MI455X (gfx1250) — compile-verified
//
#include <hip/hip_runtime.h>

#define NN 10000
#define NE 32000
#define NS 48
#define NV 16
#define NODE_DIM 96     // NS + 3*NV
#define NEF 64
#define HID 64
#define WN 4096         // NS*NS + NV*NS + NS*NV + NV*NV

typedef __attribute__((ext_vector_type(2))) float v2f;
typedef __attribute__((ext_vector_type(8))) float v8f;

__device__ __forceinline__ void atomic_add_f32(float* p, float v) {
    __hip_atomic_fetch_add(p, v, __ATOMIC_RELAXED, __HIP_MEMORY_SCOPE_AGENT);
}

// ---------------------------------------------------------------------------
// Kernel 0a: zero the output accumulator and the per-node edge counts.
// ---------------------------------------------------------------------------
__global__ void tpconv_zero_kernel(float* __restrict__ out, float* __restrict__ cnt) {
    int i = blockIdx.x * blockDim.x + threadIdx.x;
    if (i < NN * NODE_DIM) out[i] = 0.0f;
    if (i < NN) cnt[i] = 0.0f;
}

// ---------------------------------------------------------------------------
// Kernel 0b: transpose W[K][N] (row-major) -> Wt[N][K] so the WMMA B-operand
// K-pair (k, k+1) for a fixed column becomes one contiguous 8-byte load.
// ---------------------------------------------------------------------------
__global__ void tpconv_transpose_kernel(const float* __restrict__ W,
                                        float* __restrict__ Wt, int K, int N) {
    int i = blockIdx.x * blockDim.x + threadIdx.x;
    if (i >= N * K) return;
    int n = i / K, k = i - n * K;
    Wt[i] = W[(size_t)k * N + n];
}

// ---------------------------------------------------------------------------
// NT independent 16x16 (M=16 edges) fp32 GEMM tiles over K=64, each a chain
// of 16 V_WMMA_F32_16X16X4_F32, interleaved so NT accumulation chains are in
// flight simultaneously. Tiles cover contiguous columns [base, base+16*NT).
//   aop[] = A operand (16x64) in A-layout:
//     lane l: M = l%16 ; aop[kk].x -> K = 4*kk + 2*(l/16) ; .y -> K+1
//   Wt = transposed weights [N][64]: lane's B pair is one b64 load with an
//   immediate offset off a per-tile base address (1 VMEM per WMMA).
//   bias splatted into C before accumulation.
// ---------------------------------------------------------------------------
template <int NT>
__device__ __forceinline__ void tile_gemmN(const v2f* aop,
                                           const float* __restrict__ Wt,
                                           int base, int lo, int hi,
                                           const float* __restrict__ bias,
                                           v8f* c) {
#pragma unroll
    for (int n = 0; n < NT; ++n) {
        float bb = bias[base + 16 * n + lo];
        c[n] = (v8f){bb, bb, bb, bb, bb, bb, bb, bb};
    }
#pragma unroll
    for (int kk = 0; kk < 16; ++kk) {
#pragma unroll
        for (int n = 0; n < NT; ++n) {
            v2f b = *(const v2f*)(Wt + ((size_t)(base + 16 * n + lo) << 6) +
                                  4 * kk + 2 * hi);
            c[n] = __builtin_amdgcn_wmma_f32_16x16x4_f32(
                false, aop[kk], false, b, (short)0, c[n], false, false);
        }
    }
}

// ---------------------------------------------------------------------------
// Kernel 1: fused  h = relu(edge_attr@W1+b1) ;  tp_w = h@W2+b2 (tile-wise,
// never materialized) ; e3nn tensor product epilogue ; atomic scatter-add.
// One wave (32 threads) per 16 edges.
// ---------------------------------------------------------------------------
__global__ void __launch_bounds__(32)
tpconv_main_kernel(const float* __restrict__ node_attr,
                   const int* __restrict__ edge_index,
                   const float* __restrict__ edge_attr,
                   const float* __restrict__ edge_sh,
                   const float* __restrict__ w1t,     // [HID][NEF] transposed
                   const float* __restrict__ fc_b1,
                   const float* __restrict__ w2t,     // [WN][HID] transposed
                   const float* __restrict__ fc_b2,
                   float* __restrict__ out,
                   float* __restrict__ cnt) {
    __shared__ float aL[NS * 16];       // (xs*s0)/sqrt(48)      [i][e]
    __shared__ float dotL[NV * 16];     // (xv.s1)/sqrt(48)      [i][e]
    __shared__ float xsL[NS * 16];      // xs/sqrt(48)           [i][e]
    __shared__ float xvL[3][NV * 16];   // xv*s0/4               [c][i][e]
    __shared__ float s1L[3][16];        // sh[:,1:4]             [c][e]
    __shared__ float s0L[16];
    __shared__ float eaL[NEF * 16];     // edge_attr transposed  [k][e]
    __shared__ float hL[HID * 16];      // h transposed          [k][e]
    __shared__ int   srcL[16];

    const int tid = threadIdx.x;
    const int e0  = blockIdx.x * 16;
    const int lo  = tid & 15;   // N-col / M-row lane index
    const int hi  = tid >> 4;   // 0 or 1 (half-wave)
    const float inv_s48 = 0.14433756729740643f;  // 1/sqrt(48)

    // ---- per-edge sh / src indices --------------------------------------
    if (tid < 16) {
        int e = e0 + tid;
        srcL[tid]   = edge_index[e];          // edge_src
        s0L[tid]    = edge_sh[e * 4 + 0];
        s1L[0][tid] = edge_sh[e * 4 + 1];
        s1L[1][tid] = edge_sh[e * 4 + 2];
        s1L[2][tid] = edge_sh[e * 4 + 3];
    }
    __syncthreads();

    // ---- gather node_attr[edge_dst], stage scaled factors ---------------
    for (int idx = tid; idx < 16 * NODE_DIM; idx += 32) {
        int e = idx / NODE_DIM, c = idx % NODE_DIM;
        int dst = edge_index[NE + e0 + e];
        float v  = node_attr[(size_t)dst * NODE_DIM + c];
        float s0 = s0L[e];
        if (c < NS) {
            xsL[c * 16 + e] = v * inv_s48;
            aL[c * 16 + e]  = v * s0 * inv_s48;
        } else {
            int iv = (c - NS) / 3, cc = (c - NS) % 3;
            xvL[cc][iv * 16 + e] = v * s0 * 0.25f;   // 1/sqrt(16)
        }
    }
    // dot_i = xv_i . s1 (unscaled xv, reload through cache)
    for (int idx = tid; idx < 16 * NV; idx += 32) {
        int e = idx & 15, iv = idx >> 4;
        int dst = edge_index[NE + e0 + e];
        const float* p = node_attr + (size_t)dst * NODE_DIM + NS + 3 * iv;
        float d = p[0] * s1L[0][e] + p[1] * s1L[1][e] + p[2] * s1L[2][e];
        dotL[iv * 16 + e] = d * inv_s48;
    }
    // edge_attr tile, transposed for A-operand reads
    for (int idx = tid; idx < 16 * NEF; idx += 32) {
        int e = idx >> 6, k = idx & 63;
        eaL[k * 16 + e] = edge_attr[(size_t)(e0 + e) * NEF + k];
    }
    __syncthreads();

    // ---- h = relu(edge_attr @ W1 + b1): 4 chains in flight --------------
    v2f aop[16];
#pragma unroll
    for (int kk = 0; kk < 16; ++kk) {
        aop[kk].x = eaL[(4 * kk + 2 * hi) * 16 + lo];
        aop[kk].y = eaL[(4 * kk + 1 + 2 * hi) * 16 + lo];
    }
    {
        v8f hc[4];
        tile_gemmN<4>(aop, w1t, 0, lo, hi, fc_b1, hc);
#pragma unroll
        for (int nt = 0; nt < 4; ++nt)
#pragma unroll
            for (int r = 0; r < 8; ++r) {
                float v = hc[nt][r] > 0.0f ? hc[nt][r] : 0.0f;  // relu
                hL[(nt * 16 + lo) * 16 + (r + 8 * hi)] = v;     // [k][e]
            }
    }
    __syncthreads();

    // A operand for the big GEMM: h tile (16 edges x 64)
#pragma unroll
    for (int kk = 0; kk < 16; ++kk) {
        aop[kk].x = hL[(4 * kk + 2 * hi) * 16 + lo];
        aop[kk].y = hL[(4 * kk + 1 + 2 * hi) * 16 + lo];
    }

    // ---- main fused loop: tp_w tiles consumed in-register ---------------
    v8f accs0 = {}, accs1 = {}, accs2 = {};  // out_s cols 0-15,16-31,32-47
    v8f uacc  = {};                          // sum_i xs_i * w3[i][o]
    v8f vx = {}, vy = {}, vz = {};           // sum_i (xv*s0/4)[c] * w4[i][o]
    v8f c[4];

    // block 1: w1 [48][48]; one i-row = 3 tiles = 3 chains, shared factor a[i][e]
    for (int i = 0; i < NS; ++i) {
        tile_gemmN<3>(aop, w2t, i * 48, lo, hi, fc_b2, c);
        const float* f = &aL[i * 16 + 8 * hi];
#pragma unroll
        for (int r = 0; r < 8; ++r) {
            float fr = f[r];
            accs0[r] += fr * c[0][r];
            accs1[r] += fr * c[1][r];
            accs2[r] += fr * c[2][r];
        }
    }
    // block 2: w2 [16][48]; same shape, factor dot[i][e]
    for (int i = 0; i < NV; ++i) {
        tile_gemmN<3>(aop, w2t, 2304 + i * 48, lo, hi, fc_b2, c);
        const float* f = &dotL[i * 16 + 8 * hi];
#pragma unroll
        for (int r = 0; r < 8; ++r) {
            float fr = f[r];
            accs0[r] += fr * c[0][r];
            accs1[r] += fr * c[1][r];
            accs2[r] += fr * c[2][r];
        }
    }
    // block 3: w3 [48][16]; 2 tiles (= 2 i-rows) per iteration, factor xs[i][e]
    for (int t = 0; t < NS; t += 2) {
        tile_gemmN<2>(aop, w2t, 3072 + t * 16, lo, hi, fc_b2, c);
        const float* f0 = &xsL[t * 16 + 8 * hi];
        const float* f1 = &xsL[(t + 1) * 16 + 8 * hi];
#pragma unroll
        for (int r = 0; r < 8; ++r)
            uacc[r] += f0[r] * c[0][r] + f1[r] * c[1][r];
    }
    // block 4: w4 [16][16]; 2 tiles per iteration, factors (xv*s0/4)[comp][i][e]
    for (int t = 0; t < NV; t += 2) {
        tile_gemmN<2>(aop, w2t, 3840 + t * 16, lo, hi, fc_b2, c);
#pragma unroll
        for (int n = 0; n < 2; ++n) {
            const float* f0 = &xvL[0][(t + n) * 16 + 8 * hi];
            const float* f1 = &xvL[1][(t + n) * 16 + 8 * hi];
            const float* f2 = &xvL[2][(t + n) * 16 + 8 * hi];
#pragma unroll
            for (int r = 0; r < 8; ++r) {
                vx[r] += f0[r] * c[n][r];
                vy[r] += f1[r] * c[n][r];
                vz[r] += f2[r] * c[n][r];
            }
        }
    }

    // ---- scatter-add into out[edge_src] ---------------------------------
    // C-tile lane (lo,hi), vgpr r -> edge = r+8*hi, out_s col = st*16+lo
#pragma unroll
    for (int r = 0; r < 8; ++r) {
        int e = r + 8 * hi;
        size_t base = (size_t)srcL[e] * NODE_DIM;
        atomic_add_f32(out + base + 0 * 16 + lo, accs0[r]);
        atomic_add_f32(out + base + 1 * 16 + lo, accs1[r]);
        atomic_add_f32(out + base + 2 * 16 + lo, accs2[r]);
        // out_v: o = lo, component cc:  u*s1[cc] + v_cc
        size_t vb = base + NS + 3 * lo;
        atomic_add_f32(out + vb + 0, uacc[r] * s1L[0][e] + vx[r]);
        atomic_add_f32(out + vb + 1, uacc[r] * s1L[1][e] + vy[r]);
        atomic_add_f32(out + vb + 2, uacc[r] * s1L[2][e] + vz[r]);
    }
    if (tid < 16) atomic_add_f32(cnt + srcL[tid], 1.0f);
}

// ---------------------------------------------------------------------------
// Kernel 2: out = out / clip(cnt, eps) + node_attr
// ---------------------------------------------------------------------------
__global__ void tpconv_finalize_kernel(float* __restrict__ out,
                                       const float* __restrict__ cnt,
                                       const float* __restrict__ node_attr) {
    int i = blockIdx.x * blockDim.x + threadIdx.x;
    if (i >= NN * NODE_DIM) return;
    int n = i / NODE_DIM;
    float c = cnt[n];
    const float eps = 1.1920929e-7f;
    c = c < eps ? eps : c;
    out[i] = out[i] / c + node_attr[i];
}

extern "C" void kernel_launch(void* const* d_in, const int* in_sizes, int n_in,
                              void* d_out, int out_size, void* d_ws, size_t ws_size,
                              hipStream_t stream) {
    const float* node_attr  = (const float*)d_in[0];
    const int*   edge_index = (const int*)d_in[1];
    const float* edge_attr  = (const float*)d_in[2];
    const float* edge_sh    = (const float*)d_in[3];
    const float* fc_w1      = (const float*)d_in[4];
    const float* fc_b1      = (const float*)d_in[5];
    const float* fc_w2      = (const float*)d_in[6];
    const float* fc_b2      = (const float*)d_in[7];
    float* out = (float*)d_out;

    // workspace layout (floats): [cnt: 10240][w1t: 64*64][w2t: 4096*64]
    float* ws  = (float*)d_ws;
    float* cnt = ws;
    float* w1t = ws + 10240;
    float* w2t = w1t + HID * NEF;

    int ztotal = NN * NODE_DIM;
    tpconv_zero_kernel<<<(ztotal + 255) / 256, 256, 0, stream>>>(out, cnt);
    tpconv_transpose_kernel<<<(HID * NEF + 255) / 256, 256, 0, stream>>>(
        fc_w1, w1t, NEF, HID);
    tpconv_transpose_kernel<<<(WN * HID + 255) / 256, 256, 0, stream>>>(
        fc_w2, w2t, HID, WN);

    tpconv_main_kernel<<<NE / 16, 32, 0, stream>>>(
        node_attr, edge_index, edge_attr, edge_sh,
        w1t, fc_b1, w2t, fc_b2, out, cnt);

    tpconv_finalize_kernel<<<(ztotal + 255) / 256, 256, 0, stream>>>(
        out, cnt, node_attr);
}